// EConv_9457517986234
// MI455X (gfx1250) — compile-verified
//
#include <hip/hip_runtime.h>
#include <cstdint>

// EConv: out[dst] += x[src] * edge_attr   (segment scatter-add, d = 64)
//
// Roofline: edge_attr stream 256 MB (HBM) + indices 8 MB + x gathers (25.6 MB,
// L2-resident) + output atomics (25.6 MB, L2-resident) => ~12 us floor at
// 23.3 TB/s. 128 MFLOP total -> pure bandwidth/scatter problem; WMMA does not
// apply. CDNA5 paths used: global_load_async_to_lds_b128 (ASYNCcnt) to stage
// edge_attr tiles, s_wait_asynccnt, and no-return global_atomic_add_f32 at
// agent scope resolving in the 192 MB L2.

#define D        64    // feature dim
#define EPT      16    // edges per block tile  (16 edges * 64 f32 = 4 KB LDS)
#define TPB      256   // threads per block: thread t -> edge t>>4, quad t&15

__global__ void econv_zero_kernel(float4* __restrict__ out, int n4) {
    int i = blockIdx.x * blockDim.x + threadIdx.x;
    if (i < n4) out[i] = make_float4(0.f, 0.f, 0.f, 0.f);
}

__global__ __launch_bounds__(TPB) void econv_scatter_kernel(
    const float* __restrict__ x,       // [N, 64]
    const int*   __restrict__ eidx,    // [2, E]  row0 = dst, row1 = src
    const float* __restrict__ eattr,   // [E, 64]
    float*       __restrict__ out,     // [N, 64]
    int E)
{
    __shared__ __align__(16) float tile[EPT * D];   // 4 KB per block

    const int t        = threadIdx.x;
    const int tileBase = blockIdx.x * EPT;          // first edge of this tile
    const bool full    = (tileBase + EPT) <= E;

    // ---- 1) kick off async staging of the edge_attr tile into LDS -------
    // 256 lanes x 16 B = 4 KB contiguous; each lane copies its own slot and
    // later reads back exactly that slot, so a per-wave s_wait_asynccnt is
    // sufficient (no barrier).
    const float* gsrc = eattr + (size_t)tileBase * D + t * 4;
    // Generic LDS pointers carry the byte offset in the low 32 bits
    uint32_t ldsAddr = (uint32_t)(uintptr_t)(&tile[t * 4]);

    if (full) {
        asm volatile("global_load_async_to_lds_b128 %0, %1, off"
                     :: "v"(ldsAddr), "v"(gsrc)
                     : "memory");
    } else {
        // tail tile (E not multiple of EPT): plain copy of valid rows
        int e = tileBase + (t >> 4);
        if (e < E) {
            *(float4*)(&tile[t * 4]) = *(const float4*)gsrc;
        }
    }

    // ---- 2) overlap: edge indices + x gather while the copy is in flight
    const int e = tileBase + (t >> 4);   // this thread's edge
    const int q = t & 15;                // this thread's float4 within the row
    const bool active = (e < E);

    int dst = 0, src = 0;
    if (active) {
        dst = eidx[e];        // segment id (row 0)
        src = eidx[E + e];    // gather index (row 1)
    }

    float4 xv = make_float4(0.f, 0.f, 0.f, 0.f);
    if (active) {
        xv = *(const float4*)(x + (size_t)src * D + q * 4);  // L2-resident gather
    }

    // ---- 3) wait for this wave's async LDS writes, consume own slot -----
    if (full) {
        asm volatile("s_wait_asynccnt 0" ::: "memory");
    }

    if (active) {
        float4 a = *(const float4*)(&tile[t * 4]);

        float m0 = xv.x * a.x;
        float m1 = xv.y * a.y;
        float m2 = xv.z * a.z;
        float m3 = xv.w * a.w;

        float* op = out + (size_t)dst * D + q * 4;
        // relaxed agent-scope fadd -> global_atomic_add_f32 (no return);
        // destination set (25.6 MB) stays resident in the 192 MB L2
        __hip_atomic_fetch_add(op + 0, m0, __ATOMIC_RELAXED, __HIP_MEMORY_SCOPE_AGENT);
        __hip_atomic_fetch_add(op + 1, m1, __ATOMIC_RELAXED, __HIP_MEMORY_SCOPE_AGENT);
        __hip_atomic_fetch_add(op + 2, m2, __ATOMIC_RELAXED, __HIP_MEMORY_SCOPE_AGENT);
        __hip_atomic_fetch_add(op + 3, m3, __ATOMIC_RELAXED, __HIP_MEMORY_SCOPE_AGENT);
    }
}

extern "C" void kernel_launch(void* const* d_in, const int* in_sizes, int n_in,
                              void* d_out, int out_size, void* d_ws, size_t ws_size,
                              hipStream_t stream) {
    const float* x     = (const float*)d_in[0];   // [N, 64]
    const int*   eidx  = (const int*)d_in[1];     // [2, E]
    const float* eattr = (const float*)d_in[2];   // [E, 64]
    float*       out   = (float*)d_out;           // [N, 64]

    const int E = in_sizes[1] / 2;                // 1,000,000

    // 1) zero the output (harness poisons d_out with 0xAA).
    //    out_size = N*64 is divisible by 4 -> exact float4 grid, no loop.
    {
        int n4      = out_size >> 2;              // 1.6M float4
        int threads = 256;
        int blocks  = (n4 + threads - 1) / threads;
        econv_zero_kernel<<<blocks, threads, 0, stream>>>((float4*)out, n4);
    }

    // 2) scatter-add over all edges, 16 edges per 256-thread block
    {
        int blocks = (E + EPT - 1) / EPT;         // 62,500
        econv_scatter_kernel<<<blocks, TPB, 0, stream>>>(x, eidx, eattr, out, E);
    }
}